// L3MParamHead_41034117546158
// MI455X (gfx1250) — compile-verified
//
#include <hip/hip_runtime.h>
#include <hip/hip_bf16.h>

typedef __attribute__((ext_vector_type(2))) float v2f;
typedef __attribute__((ext_vector_type(8))) float v8f;

#define HID   4096
#define BATCH 16384
#define OUTC  44
#define PD    8

#define THREADS        128   // 4 wave32
#define ROWS_PER_BLOCK 64    // 4 waves x 16 rows
#define KC        64
#define XS_STRIDE 68         // padded so ds_load_b64 fragments hit all 64 banks
#define WS_STRIDE 68
#define HS_STRIDE 52
#define NPAD      48         // 44 cols padded to 3 x 16

__global__ __launch_bounds__(THREADS) void l3m_param_head_kernel(
    const float* __restrict__ x, const float* __restrict__ W,
    const float* __restrict__ bias, float* __restrict__ out)
{
    __shared__ float xs[ROWS_PER_BLOCK * XS_STRIDE]; // 4352 floats (17 KB)
    __shared__ float ws[NPAD * WS_STRIDE];           // 3264 floats (13 KB)

    const int tid  = threadIdx.x;
    const int wid  = tid >> 5;
    const int lane = tid & 31;
    const int half = lane >> 4;   // 0/1: which K pair (A/B) & which M half (C/D)
    const int l16  = lane & 15;
    const int m0   = wid * 16;    // wave's row tile inside the block
    const int rowBase = blockIdx.x * ROWS_PER_BLOCK;

    v8f acc0 = {}; v8f acc1 = {}; v8f acc2 = {};

    for (int k0 = 0; k0 < HID; k0 += KC) {
        // ---- cooperative, coalesced global -> LDS staging (float4) ----
        const float4* xg = (const float4*)x;
        const float4* wg = (const float4*)W;
        const int kq = k0 >> 2;
        #pragma unroll
        for (int i = 0; i < 8; ++i) {            // 1024 float4 of x tile (64x64)
            int f  = tid + i * THREADS;
            int r  = f >> 4;
            int c4 = f & 15;
            float4 v = xg[(size_t)(rowBase + r) * (HID / 4) + kq + c4];
            *(float4*)&xs[r * XS_STRIDE + c4 * 4] = v;
        }
        #pragma unroll
        for (int i = 0; i < 6; ++i) {            // 768 float4 of W tile (zero-padded)
            int f  = tid + i * THREADS;
            int n  = f >> 4;
            int c4 = f & 15;
            float4 v = make_float4(0.f, 0.f, 0.f, 0.f);
            if (n < OUTC)
                v = wg[(size_t)n * (HID / 4) + kq + c4];
            *(float4*)&ws[n * WS_STRIDE + c4 * 4] = v;
        }
        // deepen the pipeline: pull the K+2 tile toward L2/WGP$
        // (thread t covers row t>>1, 32-float half t&1 of the 64x64 tile)
        if (k0 + 2 * KC < HID)
            __builtin_prefetch(&x[(size_t)(rowBase + (tid >> 1)) * HID
                                  + (k0 + 2 * KC) + (tid & 1) * 32], 0, 1);
        __syncthreads();

        // ---- 16 K-steps of V_WMMA_F32_16X16X4_F32, 3 N-tiles each ----
        #pragma unroll
        for (int kk = 0; kk < KC; kk += 4) {
            const int ks = kk + 2 * half;
            // A 16x4 layout: lane L -> row L%16, VGPR0/1 = K {2*(L/16), 2*(L/16)+1}
            v2f a  = *(const v2f*)&xs[(m0 + l16) * XS_STRIDE + ks];
            // B 4x16 layout mirrors A with W_lds stored [n][k]
            v2f b0 = *(const v2f*)&ws[( 0 + l16) * WS_STRIDE + ks];
            v2f b1 = *(const v2f*)&ws[(16 + l16) * WS_STRIDE + ks];
            v2f b2 = *(const v2f*)&ws[(32 + l16) * WS_STRIDE + ks];
            acc0 = __builtin_amdgcn_wmma_f32_16x16x4_f32(false, a, false, b0, (short)0, acc0, false, false);
            acc1 = __builtin_amdgcn_wmma_f32_16x16x4_f32(false, a, false, b1, (short)0, acc1, false, false);
            acc2 = __builtin_amdgcn_wmma_f32_16x16x4_f32(false, a, false, b2, (short)0, acc2, false, false);
        }
        __syncthreads();
    }

    // ---- dump h tile to LDS (reuse xs buffer; 64*52 <= 64*68 floats) ----
    float* hs = xs;
    #pragma unroll
    for (int v = 0; v < 8; ++v) {
        // C/D layout: VGPR v -> row v (lanes 0-15) / row v+8 (lanes 16-31)
        int m = m0 + v + 8 * half;
        hs[m * HS_STRIDE +  0 + l16] = acc0[v];
        hs[m * HS_STRIDE + 16 + l16] = acc1[v];
        hs[m * HS_STRIDE + 32 + l16] = acc2[v];
    }
    __syncthreads();

    // ---- fused epilogue: one thread per row ----
    if (tid < ROWS_PER_BLOCK) {
        const int row = rowBase + tid;
        float h[OUTC];
        #pragma unroll
        for (int j = 0; j < OUTC; ++j) h[j] = hs[tid * HS_STRIDE + j] + bias[j];

        float* outP = out;
        float* outS = out + (size_t)BATCH * PD;
        float* outL = out + (size_t)BATCH * PD + (size_t)BATCH * PD * PD;

        // params = sigmoid(h[:8])
        float pr[PD];
        #pragma unroll
        for (int i = 0; i < PD; ++i) pr[i] = 1.f / (1.f + expf(-h[i]));
        #pragma unroll
        for (int i = 0; i < PD; i += 4)
            *(float4*)&outP[(size_t)row * PD + i] =
                make_float4(pr[i], pr[i + 1], pr[i + 2], pr[i + 3]);

        // sig: lower-triangular, scaled by 2/sqrt(i+1)
        float sg[PD][PD];
        #pragma unroll
        for (int i = 0; i < PD; ++i) {
            const float sc = 2.0f / sqrtf((float)(i + 1));
            float d  = h[PD + i];
            float sp = (d > 20.f) ? d : log1pf(expf(d));   // softplus
            sg[i][i] = sp * sc;
            #pragma unroll
            for (int j = 0; j < i; ++j)
                sg[i][j] = h[2 * PD + (i * (i - 1)) / 2 + j] * sc;
        }

        // logdet = log(log1p(exp(beta*p))) - log(beta), stable
        float p = 1.f;
        #pragma unroll
        for (int i = 0; i < PD; ++i) p *= sg[i][i];
        float bp  = 10000.f * p;
        float lae = (bp > 0.f) ? (bp + log1pf(expf(-bp))) : log1pf(expf(bp));
        outL[row] = logf(lae) - 9.210340371976184f;        // - log(1e4)

        // sigmas = sig . sig^T (symmetric; k <= min(i,j) = j)
        float sm[PD][PD];
        #pragma unroll
        for (int i = 0; i < PD; ++i) {
            #pragma unroll
            for (int j = 0; j <= i; ++j) {
                float s = 0.f;
                #pragma unroll
                for (int k = 0; k <= j; ++k) s += sg[i][k] * sg[j][k];
                sm[i][j] = s;
                sm[j][i] = s;
            }
        }
        #pragma unroll
        for (int i = 0; i < PD; ++i)
            #pragma unroll
            for (int j = 0; j < PD; j += 4)
                *(float4*)&outS[(size_t)row * (PD * PD) + i * PD + j] =
                    make_float4(sm[i][j], sm[i][j + 1], sm[i][j + 2], sm[i][j + 3]);
    }
}

extern "C" void kernel_launch(void* const* d_in, const int* in_sizes, int n_in,
                              void* d_out, int out_size, void* d_ws, size_t ws_size,
                              hipStream_t stream) {
    const float* x = (const float*)d_in[0];
    const float* W = (const float*)d_in[1];
    const float* b = (const float*)d_in[2];
    float* out = (float*)d_out;
    (void)in_sizes; (void)n_in; (void)out_size; (void)d_ws; (void)ws_size;

    dim3 grid(BATCH / ROWS_PER_BLOCK);  // 256 blocks -> spreads across WGPs
    dim3 block(THREADS);                // 4 wave32
    l3m_param_head_kernel<<<grid, block, 0, stream>>>(x, W, b, out);
}